// AutoCov1D_36189394436369
// MI455X (gfx1250) — compile-verified
//
#include <hip/hip_runtime.h>

// AutoCov1D on MI455X (gfx1250), wave32, WMMA f32 path + async LDS staging.
//
// cov[b,n,c] = (1/D) * sum_d p1*p2 - mean_d(p1)*mean_d(p2) + bias[c]
// p{1,2}[b,n,d,c] = sum_w w{1,2}[b,n,w] * weight[w,d,c]
// w1[b,n,w] = X[b, 8n + w], w2[b,n,w] = X[b, 8n + 64 + w]

typedef float v2f __attribute__((ext_vector_type(2)));
typedef float v8f __attribute__((ext_vector_type(8)));
typedef int   v4i __attribute__((ext_vector_type(4)));

#define WSZ   64      // window size (K of the GEMM)
#define DDIM  32      // d dimension (reduced by covariance)
#define CDIM  128     // channels
#define CPAD  132     // padded LDS row stride: half-waves land 8 banks apart
#define NWIN  497     // windows per batch
#define TLEN  4096    // sequence length
#define NT    16      // windows (GEMM M) per tile
#define XSEG  248     // X floats per window tile: 15*8 + 64 + 64

#if defined(__AMDGCN__) && __has_builtin(__builtin_amdgcn_global_load_async_to_lds_b128)
#define HAVE_ASYNC_LDS 1
#else
#define HAVE_ASYNC_LDS 0
#endif

#define GLOBAL_V4I(p) ((__attribute__((address_space(1))) v4i*)(p))
#define LDS_V4I(p)    ((__attribute__((address_space(3))) v4i*)(p))

// Stage weight[:, d, :] (64 x 128 f32) into an LDS buffer with row stride CPAD.
// Async path: 8 x b128 DMA ops per thread, tracked by ASYNCcnt (no VGPR round-trip).
__device__ __forceinline__ void stage_weight_async(const float* __restrict__ slice, // Wt + d*CDIM
                                                   float* lds_dst, int t)
{
#if HAVE_ASYNC_LDS
    #pragma unroll
    for (int i = 0; i < 8; ++i) {
        const int e = (i * 256 + t) * 4;          // element index in the 64x128 slice
        const int w = e >> 7;
        const int c = e & 127;                    // multiple of 4 -> b128 stays in-row
        __builtin_amdgcn_global_load_async_to_lds_b128(
            GLOBAL_V4I(slice + w * (DDIM * CDIM) + c),
            LDS_V4I(lds_dst + w * CPAD + c),
            0, 0);
    }
#else
    (void)slice; (void)lds_dst; (void)t;
#endif
}

__device__ __forceinline__ void stage_weight_sync(const float* __restrict__ slice,
                                                  float* lds_dst, int t)
{
    #pragma unroll 4
    for (int i = 0; i < 32; ++i) {
        const int e = i * 256 + t;
        const int w = e >> 7;
        const int c = e & 127;
        lds_dst[w * CPAD + c] = slice[w * (DDIM * CDIM) + c];
    }
}

__global__ __launch_bounds__(256)
void autocov1d_wmma_f32(const float* __restrict__ X,
                        const float* __restrict__ Wt,    // [64][32][128]
                        const float* __restrict__ bias,  // [128]
                        float* __restrict__ out)         // [32][497][128]
{
    __shared__ float lds_w[2][WSZ * CPAD];  // double-buffered weight slice (2 x 33 KB)
    __shared__ float lds_x[XSEG];           // X segment for this window tile

    const int t    = threadIdx.x;
    const int lane = t & 31;
    const int wave = t >> 5;                // 0..7 -> c-tile
    const int c0   = wave * 16;
    const int b    = blockIdx.y;
    const int n0   = blockIdx.x * NT;

#if HAVE_ASYNC_LDS
    stage_weight_async(Wt, lds_w[0], t);    // prefetch d = 0 while we stage X
#endif

    // ---- stage X segment (zero-padded past T for the ragged last tile) ----
    if (t < XSEG) {
        const int g = n0 * 8 + t;
        lds_x[t] = (g < TLEN) ? X[b * TLEN + g] : 0.0f;
    }
#if HAVE_ASYNC_LDS
    asm volatile("s_wait_asynccnt 0x0" ::: "memory");
#endif
    __syncthreads();

    // ---- build A fragments for WMMA_F32_16X16X4_F32 ----
    // A tile is 16(M) x 4(K). Lane l: M = l&15; VGPR j holds K = 2*(l>>4) + j.
    // a[k], k=0..15 -> w1 (w = 4k..4k+3); a[k+16] -> w2 (w2 = w1 + 64 = 4*(k+16)).
    const int row  = lane & 15;
    const int koff = (lane >> 4) * 2;
    v2f a[32];
    #pragma unroll
    for (int k = 0; k < 32; ++k) {
        const int wi = row * 8 + k * 4 + koff;
        a[k].x = lds_x[wi];
        a[k].y = lds_x[wi + 1];
    }

    v8f S1 = {}, S2 = {}, Sp = {};
    const int nn = lane & 15;               // GEMM N index within the c-tile

    for (int d = 0; d < DDIM; ++d) {
#if HAVE_ASYNC_LDS
        const float* wbuf = lds_w[d & 1];
        if (d + 1 < DDIM)                   // overlap next slice DMA with this d's WMMAs
            stage_weight_async(Wt + (d + 1) * CDIM, lds_w[(d + 1) & 1], t);
#else
        const float* wbuf = lds_w[0];
        __syncthreads();                    // previous slice fully consumed
        stage_weight_sync(Wt + d * CDIM, lds_w[0], t);
        __syncthreads();
#endif

        // ---- two 16x16x64 GEMMs as 16 chained f32 WMMAs each ----
        v8f acc1 = {}, acc2 = {};
        #pragma unroll
        for (int k = 0; k < 16; ++k) {
            const int wrow = k * 4 + koff;  // B: K row pair base
            v2f bf;
            bf.x = wbuf[wrow * CPAD + c0 + nn];
            bf.y = wbuf[(wrow + 1) * CPAD + c0 + nn];
            acc1 = __builtin_amdgcn_wmma_f32_16x16x4_f32(
                       false, a[k],      false, bf, (short)0, acc1, false, false);
            acc2 = __builtin_amdgcn_wmma_f32_16x16x4_f32(
                       false, a[k + 16], false, bf, (short)0, acc2, false, false);
        }
        S1 += acc1;                 // sum_d p1
        S2 += acc2;                 // sum_d p2
        Sp += acc1 * acc2;          // sum_d p1*p2

#if HAVE_ASYNC_LDS
        asm volatile("s_wait_asynccnt 0x0" ::: "memory");
        __syncthreads();            // next slice resident + this slice's readers done
#endif
    }

    // ---- epilogue: cov = Sp/D - (S1/D)(S2/D) + bias, masked store ----
    const float bv   = bias[c0 + nn];
    const float invD = 1.0f / (float)DDIM;
    #pragma unroll
    for (int r = 0; r < 8; ++r) {
        const int m = r + ((lane >> 4) * 8);   // C/D layout: M = r + 8*(l>=16)
        const int n = n0 + m;
        if (n < NWIN) {
            const float cov = Sp[r] * invD - (S1[r] * invD) * (S2[r] * invD) + bv;
            out[((size_t)b * NWIN + n) * CDIM + c0 + nn] = cov;
        }
    }
}

extern "C" void kernel_launch(void* const* d_in, const int* in_sizes, int n_in,
                              void* d_out, int out_size, void* d_ws, size_t ws_size,
                              hipStream_t stream) {
    const float* X    = (const float*)d_in[0];   // (32, 4096)
    const float* Wt   = (const float*)d_in[1];   // (64, 32, 128)
    const float* bias = (const float*)d_in[2];   // (128,)
    float* out        = (float*)d_out;           // (32, 497, 128)

    dim3 grid(32, 32);    // 32 window tiles (512 >= 497), 32 batches
    dim3 block(256);      // 8 wave32 -> 8 c-tiles of 16 channels
    autocov1d_wmma_f32<<<grid, block, 0, stream>>>(X, Wt, bias, out);
}